// SinkhornRankSort_89464168775714
// MI455X (gfx1250) — compile-verified
//
#include <hip/hip_runtime.h>
#include <math.h>

#define N 2048
#define EPSF 0.01f
#define INV_EPS 100.0f
#define NITER 20
#define LOG_N 7.6246189861f   /* ln(2048) = 11*ln2 */
#define ELPL (N / 32)         /* elements per lane in softmin reductions */

typedef __attribute__((ext_vector_type(2))) float v2f;
typedef __attribute__((ext_vector_type(8))) float v8f;

__device__ __forceinline__ float wave_min32(float v) {
    #pragma unroll
    for (int m = 16; m >= 1; m >>= 1) v = fminf(v, __shfl_xor(v, m, 32));
    return v;
}
__device__ __forceinline__ float wave_sum32(float v) {
    #pragma unroll
    for (int m = 16; m >= 1; m >>= 1) v += __shfl_xor(v, m, 32);
    return v;
}

// ---------------------------------------------------------------------------
// Phase 1: standardize one batch row; also zero alpha/beta for the Sinkhorn.
//   Xt = sigmoid( normalize(X - sum(X)) * sqrt(N) )     (faithful: SUM, not mean)
// ---------------------------------------------------------------------------
__global__ __launch_bounds__(256) void k_standardize(const float* __restrict__ X,
                                                     float* __restrict__ Xt,
                                                     float* __restrict__ alpha,
                                                     float* __restrict__ beta) {
    const int b = blockIdx.x;
    const float* x = X + (size_t)b * N;
    __shared__ float red[8];
    const int tid = threadIdx.x;

    float s = 0.f;
    for (int i = tid; i < N; i += 256) s += x[i];
    s = wave_sum32(s);
    if ((tid & 31) == 0) red[tid >> 5] = s;
    __syncthreads();
    float total = 0.f;
    #pragma unroll
    for (int w = 0; w < 8; ++w) total += red[w];
    __syncthreads();

    float ss = 0.f;
    for (int i = tid; i < N; i += 256) { float t = x[i] - total; ss += t * t; }
    ss = wave_sum32(ss);
    if ((tid & 31) == 0) red[tid >> 5] = ss;
    __syncthreads();
    float nrm2 = 0.f;
    #pragma unroll
    for (int w = 0; w < 8; ++w) nrm2 += red[w];

    const float nrm   = sqrtf(nrm2);
    const float scale = sqrtf((float)N) / fmaxf(nrm, 1e-12f);
    for (int i = tid; i < N; i += 256) {
        float t = (x[i] - total) * scale;              // TAU = 1
        Xt[(size_t)b * N + i]    = 1.0f / (1.0f + __expf(-t));
        alpha[(size_t)b * N + i] = 0.f;
        beta[(size_t)b * N + i]  = 0.f;
    }
}

// ---------------------------------------------------------------------------
// Half-iteration: beta_j = -eps*ln(n) + softmin_i( (x_i - y_j)^2 - alpha_i )
// (beta_j's own value cancels inside the softmin -> independent update)
// One wave per output j; per-lane VGPR cache of the 64 partial values.
// ---------------------------------------------------------------------------
__global__ __launch_bounds__(256) void k_beta(const float* __restrict__ Xt,
                                              const float* __restrict__ alpha,
                                              float* __restrict__ beta) {
    const int b = blockIdx.y;
    __shared__ float xs[N];
    __shared__ float as[N];
    const int tid = threadIdx.x;
    for (int i = tid; i < N; i += 256) {
        xs[i] = Xt[(size_t)b * N + i];
        as[i] = alpha[(size_t)b * N + i];
    }
    __syncthreads();

    const int lane = tid & 31, wave = tid >> 5;
    const int j = blockIdx.x * 8 + wave;
    const float y = (float)j * (1.0f / (float)(N - 1));

    float v[ELPL];
    float m = 3.4e38f;
    #pragma unroll
    for (int t = 0; t < ELPL; ++t) {
        const int i = lane + 32 * t;
        const float d = xs[i] - y;
        const float val = __builtin_fmaf(d, d, -as[i]);
        v[t] = val;
        m = fminf(m, val);
    }
    m = wave_min32(m);
    float s = 0.f;
    #pragma unroll
    for (int t = 0; t < ELPL; ++t) s += __expf((m - v[t]) * INV_EPS);
    s = wave_sum32(s);
    if (lane == 0) beta[(size_t)b * N + j] = m - EPSF * (__logf(s) + LOG_N);
}

// ---------------------------------------------------------------------------
// Half-iteration: alpha_i = -eps*ln(n) + softmin_j( (x_i - y_j)^2 - beta_j )
// ---------------------------------------------------------------------------
__global__ __launch_bounds__(256) void k_alpha(const float* __restrict__ Xt,
                                               float* __restrict__ alpha,
                                               const float* __restrict__ beta) {
    const int b = blockIdx.y;
    __shared__ float bs[N];
    const int tid = threadIdx.x;
    for (int i = tid; i < N; i += 256) bs[i] = beta[(size_t)b * N + i];
    __syncthreads();

    const int lane = tid & 31, wave = tid >> 5;
    const int i = blockIdx.x * 8 + wave;
    const float xi = Xt[(size_t)b * N + i];
    const float invn1 = 1.0f / (float)(N - 1);

    float v[ELPL];
    float m = 3.4e38f;
    #pragma unroll
    for (int t = 0; t < ELPL; ++t) {
        const int j = lane + 32 * t;
        const float d = xi - (float)j * invn1;
        const float val = __builtin_fmaf(d, d, -bs[j]);
        v[t] = val;
        m = fminf(m, val);
    }
    m = wave_min32(m);
    float s = 0.f;
    #pragma unroll
    for (int t = 0; t < ELPL; ++t) s += __expf((m - v[t]) * INV_EPS);
    s = wave_sum32(s);
    if (lane == 0) alpha[(size_t)b * N + i] = m - EPSF * (__logf(s) + LOG_N);
}

// ---------------------------------------------------------------------------
// Final pass A (M = j orientation): one wave owns 16 columns j0..j0+15,
// streams i in chunks of 4. Each lane produces p in the WMMA 16x4 f32
// A-operand layout (lanes 0-15: K=0,1 ; lanes 16-31: K=2,3), writes P
// coalesced, and accumulates  S_j = sum_i P_ij * X_i  on the matrix pipe:
//   A[m][k] = P[ic+k][j0+m],  B[k][n] = X[ic+k]  =>  D[m][*] = partial S.
// ---------------------------------------------------------------------------
__global__ __launch_bounds__(256) void k_final_SP(const float* __restrict__ Xt,
                                                  const float* __restrict__ Xraw,
                                                  const float* __restrict__ alpha,
                                                  const float* __restrict__ beta,
                                                  float* __restrict__ P,
                                                  float* __restrict__ S) {
    const int b = blockIdx.y;
    __shared__ float xs[N];   // Xt row (for cost)
    __shared__ float as[N];   // alpha row
    __shared__ float xr[N];   // raw X row (for S)
    const int tid = threadIdx.x;
    for (int i = tid; i < N; i += 256) {
        xs[i] = Xt[(size_t)b * N + i];
        as[i] = alpha[(size_t)b * N + i];
        xr[i] = Xraw[(size_t)b * N + i];
    }
    __syncthreads();

    const int lane = tid & 31, wave = tid >> 5;
    const int j0 = (blockIdx.x * 8 + wave) * 16;
    const int jl = j0 + (lane & 15);
    const float yj = (float)jl * (1.0f / (float)(N - 1));
    const float bj = beta[(size_t)b * N + jl];
    const int khi = (lane >> 4) << 1;           // 0 for lanes 0-15, 2 for 16-31
    float* __restrict__ Pb = P + (size_t)b * N * N;

    v8f acc = {};
    for (int ic = 0; ic < N; ic += 4) {
        const int i0 = ic + khi;
        const int i1 = i0 + 1;
        const float d0 = xs[i0] - yj;
        const float d1 = xs[i1] - yj;
        const float p0 = __expf((as[i0] + bj - d0 * d0) * INV_EPS);
        const float p1 = __expf((as[i1] + bj - d1 * d1) * INV_EPS);
        Pb[(size_t)i0 * N + jl] = p0;           // coalesced across lanes
        Pb[(size_t)i1 * N + jl] = p1;
        v2f a;  a.x = p0;        a.y = p1;       // A-matrix 16x4 layout
        v2f bm; bm.x = xr[i0];   bm.y = xr[i1];  // B[k][n] broadcast of X
        acc = __builtin_amdgcn_wmma_f32_16x16x4_f32(false, a, false, bm,
                                                    (short)0, acc, false, false);
    }
    // D layout: VGPR r, lanes 0-15 -> M=r ; lanes 16-31 -> M=8+r ; take N=0
    const float nf = (float)N;
    if (lane == 0) {
        #pragma unroll
        for (int r = 0; r < 8; ++r) S[(size_t)b * N + j0 + r] = acc[r] * nf;
    }
    if (lane == 16) {
        #pragma unroll
        for (int r = 0; r < 8; ++r) S[(size_t)b * N + j0 + 8 + r] = acc[r] * nf;
    }
}

// ---------------------------------------------------------------------------
// Final pass B (M = i orientation): one wave owns 16 rows, streams j in
// chunks of 4, accumulating  R_i = sum_j P_ij * b_bar_j  via WMMA:
//   A[m][k] = P[i0+m][jc+k],  B[k][n] = b_bar[jc+k].
// (p recomputed: 33.5M extra exps, ~2.5% of the Sinkhorn exp work, beats
//  cross-lane atomics.)
// ---------------------------------------------------------------------------
__global__ __launch_bounds__(256) void k_final_R(const float* __restrict__ Xt,
                                                 const float* __restrict__ alpha,
                                                 const float* __restrict__ beta,
                                                 float* __restrict__ R) {
    const int b = blockIdx.y;
    __shared__ float bs[N];
    const int tid = threadIdx.x;
    for (int i = tid; i < N; i += 256) bs[i] = beta[(size_t)b * N + i];
    __syncthreads();

    const int lane = tid & 31, wave = tid >> 5;
    const int i0 = (blockIdx.x * 8 + wave) * 16;
    const int il = i0 + (lane & 15);
    const float xi = Xt[(size_t)b * N + il];
    const float ai = alpha[(size_t)b * N + il];
    const int khi = (lane >> 4) << 1;
    const float invn1 = 1.0f / (float)(N - 1);
    const float invn  = 1.0f / (float)N;

    v8f acc = {};
    for (int jc = 0; jc < N; jc += 4) {
        const int j0 = jc + khi;
        const int j1 = j0 + 1;
        const float d0 = xi - (float)j0 * invn1;
        const float d1 = xi - (float)j1 * invn1;
        const float p0 = __expf((ai + bs[j0] - d0 * d0) * INV_EPS);
        const float p1 = __expf((ai + bs[j1] - d1 * d1) * INV_EPS);
        v2f a;  a.x = p0;                       a.y = p1;
        v2f bm; bm.x = (float)(1 + j0) * invn;  bm.y = (float)(1 + j1) * invn;
        acc = __builtin_amdgcn_wmma_f32_16x16x4_f32(false, a, false, bm,
                                                    (short)0, acc, false, false);
    }
    const float n2 = (float)N * (float)N;
    if (lane == 0) {
        #pragma unroll
        for (int r = 0; r < 8; ++r) R[(size_t)b * N + i0 + r] = acc[r] * n2;
    }
    if (lane == 16) {
        #pragma unroll
        for (int r = 0; r < 8; ++r) R[(size_t)b * N + i0 + 8 + r] = acc[r] * n2;
    }
}

// ---------------------------------------------------------------------------
extern "C" void kernel_launch(void* const* d_in, const int* in_sizes, int n_in,
                              void* d_out, int out_size, void* d_ws, size_t ws_size,
                              hipStream_t stream) {
    (void)n_in; (void)out_size; (void)ws_size;
    const float* X = (const float*)d_in[0];
    const int B = in_sizes[0] / N;              // 8

    float* ws    = (float*)d_ws;                // 3 * B * N floats (~196 KB)
    float* Xt    = ws;
    float* alpha = ws + (size_t)B * N;
    float* beta  = ws + (size_t)2 * B * N;

    float* out = (float*)d_out;                 // [P | R | S]
    float* P = out;
    float* R = out + (size_t)B * N * N;
    float* S = R + (size_t)B * N;

    k_standardize<<<B, 256, 0, stream>>>(X, Xt, alpha, beta);

    for (int it = 0; it < NITER; ++it) {
        k_beta <<<dim3(N / 8, B), 256, 0, stream>>>(Xt, alpha, beta);
        k_alpha<<<dim3(N / 8, B), 256, 0, stream>>>(Xt, alpha, beta);
    }

    k_final_SP<<<dim3(N / 16 / 8, B), 256, 0, stream>>>(Xt, X, alpha, beta, P, S);
    k_final_R <<<dim3(N / 16 / 8, B), 256, 0, stream>>>(Xt, alpha, beta, R);
}